// TaskAlignedAssigner_86028194939326
// MI455X (gfx1250) — compile-verified
//
#include <hip/hip_runtime.h>
#include <hip/hip_bf16.h>

// ---------------- problem constants (from reference setup_inputs) ------------
#define BS    32
#define NA    8400
#define NMAX  32
#define NC    80
#define TPK   13
#define AT    525          // NA / 16 anchor tiles
#define FEPS  1e-9f        // EPS
#define IEPS  1e-7f        // IOU_EPS
#define INV_PI2_4 0.40528473456935108577f  // 4 / pi^2

typedef __attribute__((ext_vector_type(2))) float v2f;
typedef __attribute__((ext_vector_type(8))) float v8f;

// =============================================================================
// K1: fused one-hot GEMM (WMMA f32 16x16x4, exact) + in-gt mask + CIoU
//     -> align_metric[b][m][a], overlaps[b][m][a]
// one wave (32 threads) per 16-anchor tile; BOTH 16-gt tiles per wave so
// pd_scores (86 MB, the dominant read) is streamed exactly once.
// =============================================================================
__global__ __launch_bounds__(32) void k_metric(
    const float* __restrict__ pd_scores,   // (BS, NA, NC)
    const float* __restrict__ pd_bboxes,   // (BS, NA, 4)
    const float* __restrict__ anc,         // (NA, 2)
    const int*   __restrict__ gt_labels,   // (BS, NMAX)
    const float* __restrict__ gt_bboxes,   // (BS, NMAX, 4)
    const float* __restrict__ mask_gt,     // (BS, NMAX)
    float* __restrict__ align_m,           // (BS, NMAX, NA)
    float* __restrict__ ovl)               // (BS, NMAX, NA)
{
  const int tile = blockIdx.x;          // 0 .. BS*AT-1
  const int b    = tile / AT;
  const int at   = tile - b * AT;

  const int l   = threadIdx.x;          // lane 0..31
  const int hi  = (l >= 16) ? 1 : 0;
  const int m_l = hi ? (l - 16) : l;    // A-matrix row owned by this lane
  const int n_l = l & 15;               // B/D column owned by this lane
  const int a   = at * 16 + n_l;        // anchor index for this lane

  const int label0 = gt_labels[b * NMAX + m_l];        // gt tile 0 (m 0..15)
  const int label1 = gt_labels[b * NMAX + 16 + m_l];   // gt tile 1 (m 16..31)
  const float* bp  = pd_scores + ((size_t)(b * NA + a)) * NC;

  // prefetch next anchor tile's score row (global_prefetch_b8)
  if (at + 1 < AT) __builtin_prefetch(bp + 16 * NC, 0, 1);

  // D = onehot(labels) x pd_scores^T, K=80 in steps of 4; B fragment shared
  // by both gt tiles -> one pd_scores read feeds two WMMAs.
  v8f acc0 = {}, acc1 = {};
#pragma unroll
  for (int k = 0; k < NC; k += 4) {
    const int kk = k + (hi ? 2 : 0);
    v2f B;  B.x = bp[kk];  B.y = bp[kk + 1];
    v2f A0, A1;
    A0.x = (label0 == kk)     ? 1.0f : 0.0f;
    A0.y = (label0 == kk + 1) ? 1.0f : 0.0f;
    A1.x = (label1 == kk)     ? 1.0f : 0.0f;
    A1.y = (label1 == kk + 1) ? 1.0f : 0.0f;
    acc0 = __builtin_amdgcn_wmma_f32_16x16x4_f32(
        false, A0, false, B, (short)0, acc0, false, false);
    acc1 = __builtin_amdgcn_wmma_f32_16x16x4_f32(
        false, A1, false, B, (short)0, acc1, false, false);
  }

  // per-lane anchor data (column fixed per lane)
  const float ax = anc[a * 2 + 0];
  const float ay = anc[a * 2 + 1];
  const float* pb = pd_bboxes + ((size_t)(b * NA + a)) * 4;
  const float px1 = pb[0], py1 = pb[1], px2 = pb[2], py2 = pb[3];

#pragma unroll
  for (int mt = 0; mt < 2; ++mt) {
#pragma unroll
    for (int v = 0; v < 8; ++v) {
      const int m = mt * 16 + v + (hi ? 8 : 0);   // D layout: lanes 16-31 -> M=v+8
      const float sc = (mt == 0) ? acc0[v] : acc1[v];
      const float* g = gt_bboxes + ((size_t)(b * NMAX + m)) * 4;
      const float lx = g[0], ly = g[1], rx = g[2], ry = g[3];

      // mask_in_gts * mask_gt
      float dmin = fminf(fminf(ax - lx, ay - ly), fminf(rx - ax, ry - ay));
      float maskf = (dmin > FEPS) ? mask_gt[b * NMAX + m] : 0.0f;

      // CIoU(gt, pd) exactly as reference
      float w1 = rx - lx,  h1 = ry - ly + IEPS;
      float w2 = px2 - px1, h2 = py2 - py1 + IEPS;
      float iw = fmaxf(fminf(rx, px2) - fmaxf(lx, px1), 0.0f);
      float ih = fmaxf(fminf(ry, py2) - fmaxf(ly, py1), 0.0f);
      float inter = iw * ih;
      float uni = w1 * h1 + w2 * h2 - inter + IEPS;
      float iou = inter / uni;
      float cw = fmaxf(rx, px2) - fminf(lx, px1);
      float ch = fmaxf(ry, py2) - fminf(ly, py1);
      float c2 = cw * cw + ch * ch + IEPS;
      float dx = px1 + px2 - lx - rx;
      float dy = py1 + py2 - ly - ry;
      float rho2 = (dx * dx + dy * dy) * 0.25f;
      float dat = atanf(w2 / h2) - atanf(w1 / h1);
      float vv = INV_PI2_4 * dat * dat;
      float alpha = vv / (vv - iou + 1.0f + IEPS);
      float ciou = iou - (rho2 / c2 + vv * alpha);

      float overlap = fmaxf(ciou, 0.0f) * maskf;
      float score   = sc * maskf;               // bbox_scores ** ALPHA (ALPHA=1)
      float o2 = overlap * overlap;
      float am = score * (o2 * o2 * o2);        // overlaps ** 6

      size_t off = (size_t)(b * NMAX + m) * NA + a;
      align_m[off] = am;
      ovl[off]     = overlap;
    }
  }
}

// sorted-13 insertion (descending value, ascending index tie-break = jax top_k)
__device__ __forceinline__ void topk_insert(float (&lv)[TPK], int (&li)[TPK],
                                            float v, int a) {
  if (v > lv[TPK - 1] || (v == lv[TPK - 1] && a < li[TPK - 1])) {
    int p = TPK - 1;
    while (p > 0 && (v > lv[p - 1] || (v == lv[p - 1] && a < li[p - 1]))) {
      lv[p] = lv[p - 1]; li[p] = li[p - 1]; --p;
    }
    lv[p] = v; li[p] = a;
  }
}

// =============================================================================
// K2: per (b,m) row top-13 over na=8400; zero + scatter into dense mask_pos
// one 256-thread block per row
// =============================================================================
__global__ __launch_bounds__(256) void k_topk(
    const float* __restrict__ align_m,
    const float* __restrict__ anc,
    const float* __restrict__ gt_bboxes,
    const float* __restrict__ mask_gt,
    float* __restrict__ mask_pos)
{
  const int row = blockIdx.x;                 // b*NMAX + m
  const float* r = align_m + (size_t)row * NA;
  float* mp = mask_pos + (size_t)row * NA;
  const int t = threadIdx.x;

  for (int a = t; a < NA; a += 256) mp[a] = 0.0f;

  float lv[TPK]; int li[TPK];
#pragma unroll
  for (int i = 0; i < TPK; ++i) { lv[i] = -INFINITY; li[i] = 0x7fffffff; }
  for (int a = t; a < NA; a += 256) topk_insert(lv, li, r[a], a);

  __shared__ float sv[256 * TPK];
  __shared__ int   si[256 * TPK];
#pragma unroll
  for (int i = 0; i < TPK; ++i) { sv[t * TPK + i] = lv[i]; si[t * TPK + i] = li[i]; }
  __syncthreads();

  if (t < 32) {                               // first wave merges 3328 candidates
#pragma unroll
    for (int i = 0; i < TPK; ++i) { lv[i] = -INFINITY; li[i] = 0x7fffffff; }
    for (int c = t; c < 256 * TPK; c += 32) topk_insert(lv, li, sv[c], si[c]);

    int pos = 0;
    int mysel = -1;
    for (int sel = 0; sel < TPK; ++sel) {
      float bv = (pos < TPK) ? lv[pos] : -INFINITY;
      int   bi = (pos < TPK) ? li[pos] : 0x7fffffff;
      for (int off = 16; off; off >>= 1) {    // wave32 max-reduce (val, idx)
        float ov = __shfl_xor(bv, off, 32);
        int   oi = __shfl_xor(bi, off, 32);
        if (ov > bv || (ov == bv && oi < bi)) { bv = ov; bi = oi; }
      }
      if (pos < TPK && li[pos] == bi) ++pos;  // anchor indices unique -> one lane pops
      if (t == sel) mysel = bi;
    }

    if (t < TPK && mysel >= 0 && mysel < NA) {
      // mask_pos = topk * mask_in_gts * mask_gt
      const float* g = gt_bboxes + (size_t)row * 4;
      float ax = anc[mysel * 2 + 0], ay = anc[mysel * 2 + 1];
      float dmin = fminf(fminf(ax - g[0], ay - g[1]),
                         fminf(g[2] - ax, g[3] - ay));
      mp[mysel] = (dmin > FEPS) ? mask_gt[row] : 0.0f;
    }
  }
}

// =============================================================================
// K3: per (b,a) column: fg count, multi-positive resolution via argmax(overlaps),
//     emit target_labels / target_bboxes / fg_mask / target_gt_idx
// =============================================================================
__global__ __launch_bounds__(256) void k_resolve(
    float* __restrict__ mask_pos,
    const float* __restrict__ ovl,
    const int*   __restrict__ gt_labels,
    const float* __restrict__ gt_bboxes,
    float* __restrict__ out_labels,
    float* __restrict__ out_bboxes,
    float* __restrict__ out_fg,
    float* __restrict__ out_idx)
{
  const int idx = blockIdx.x * 256 + threadIdx.x;
  if (idx >= BS * NA) return;
  const int b = idx / NA, a = idx - b * NA;

  float fg = 0.0f; int first = -1;
#pragma unroll
  for (int m = 0; m < NMAX; ++m) {
    float v = mask_pos[(size_t)(b * NMAX + m) * NA + a];
    fg += v;
    if (first < 0 && v > 0.0f) first = m;
  }

  int tgt; float fgf;
  if (fg > 1.0f) {
    float bestv = -INFINITY; int bm = 0;
#pragma unroll
    for (int m = 0; m < NMAX; ++m) {          // first-occurrence argmax (jnp.argmax)
      float o = ovl[(size_t)(b * NMAX + m) * NA + a];
      if (o > bestv) { bestv = o; bm = m; }
    }
#pragma unroll
    for (int m = 0; m < NMAX; ++m)
      mask_pos[(size_t)(b * NMAX + m) * NA + a] = (m == bm) ? 1.0f : 0.0f;
    tgt = bm; fgf = 1.0f;
  } else {
    tgt = (first >= 0) ? first : 0;
    fgf = fg;
  }

  int lbl = gt_labels[b * NMAX + tgt]; if (lbl < 0) lbl = 0;
  out_labels[idx] = (float)lbl;
  out_idx[idx]    = (float)tgt;
  out_fg[idx]     = (fgf > 0.0f) ? 1.0f : 0.0f;
  const float* g = gt_bboxes + (size_t)(b * NMAX + tgt) * 4;
  float* ob = out_bboxes + (size_t)idx * 4;
  ob[0] = g[0]; ob[1] = g[1]; ob[2] = g[2]; ob[3] = g[3];
}

// =============================================================================
// K4: per-row maxima of masked align / overlaps (wave32 reduction)
// =============================================================================
__global__ __launch_bounds__(32) void k_rowmax(
    const float* __restrict__ align_m,
    const float* __restrict__ ovl,
    const float* __restrict__ mask_pos,
    float* __restrict__ pos_align,
    float* __restrict__ pos_over)
{
  const int row = blockIdx.x;
  const size_t base = (size_t)row * NA;
  float ma = 0.0f, mo = 0.0f;                  // all candidates are >= 0
  for (int a = threadIdx.x; a < NA; a += 32) {
    float mp = mask_pos[base + a];
    ma = fmaxf(ma, align_m[base + a] * mp);
    mo = fmaxf(mo, ovl[base + a] * mp);
  }
  for (int off = 16; off; off >>= 1) {
    ma = fmaxf(ma, __shfl_xor(ma, off, 32));
    mo = fmaxf(mo, __shfl_xor(mo, off, 32));
  }
  if (threadIdx.x == 0) { pos_align[row] = ma; pos_over[row] = mo; }
}

// =============================================================================
// K5: norm per (b,a), scatter one-hot * norm into target_scores (BS,NA,NC)
// =============================================================================
__global__ __launch_bounds__(256) void k_scores(
    const float* __restrict__ align_m,
    const float* __restrict__ mask_pos,
    const float* __restrict__ pos_align,
    const float* __restrict__ pos_over,
    const float* __restrict__ out_labels,
    const float* __restrict__ out_fg,
    float* __restrict__ out_scores)
{
  const int idx = blockIdx.x * 256 + threadIdx.x;
  if (idx >= BS * NA) return;
  const int b = idx / NA, a = idx - b * NA;

  float norm = 0.0f;
#pragma unroll
  for (int m = 0; m < NMAX; ++m) {
    const int r = b * NMAX + m;
    float mp  = mask_pos[(size_t)r * NA + a];
    float val = align_m[(size_t)r * NA + a] * mp * pos_over[r] / (pos_align[r] + FEPS);
    norm = fmaxf(norm, val);
  }

  const int lbl = (int)out_labels[idx];
  const float fg = out_fg[idx];
  float* os = out_scores + (size_t)idx * NC;
#pragma unroll
  for (int c = 0; c < NC; ++c) os[c] = 0.0f;
  if (fg > 0.0f && lbl >= 0 && lbl < NC) os[lbl] = norm;
}

// =============================================================================
extern "C" void kernel_launch(void* const* d_in, const int* in_sizes, int n_in,
                              void* d_out, int out_size, void* d_ws, size_t ws_size,
                              hipStream_t stream) {
  const float* pd_scores = (const float*)d_in[0];   // (32, 8400, 80)
  const float* pd_bboxes = (const float*)d_in[1];   // (32, 8400, 4)
  const float* anc       = (const float*)d_in[2];   // (8400, 2)
  const int*   gt_labels = (const int*)  d_in[3];   // (32, 32, 1)
  const float* gt_bboxes = (const float*)d_in[4];   // (32, 32, 4)
  const float* mask_gt   = (const float*)d_in[5];   // (32, 32, 1)

  // workspace layout
  const size_t A = (size_t)BS * NMAX * NA;          // 8,601,600 elems
  float* align_m   = (float*)d_ws;
  float* ovl       = align_m + A;
  float* mask_pos  = ovl + A;
  float* pos_align = mask_pos + A;
  float* pos_over  = pos_align + BS * NMAX;

  // output layout: labels | bboxes | scores | fg | gt_idx (flat, float32)
  float* out        = (float*)d_out;
  float* out_labels = out;
  float* out_bboxes = out_labels + (size_t)BS * NA;
  float* out_scores = out_bboxes + (size_t)BS * NA * 4;
  float* out_fg     = out_scores + (size_t)BS * NA * NC;
  float* out_idx    = out_fg     + (size_t)BS * NA;

  k_metric<<<BS * AT, 32, 0, stream>>>(pd_scores, pd_bboxes, anc, gt_labels,
                                       gt_bboxes, mask_gt, align_m, ovl);
  k_topk<<<BS * NMAX, 256, 0, stream>>>(align_m, anc, gt_bboxes, mask_gt, mask_pos);
  k_resolve<<<(BS * NA + 255) / 256, 256, 0, stream>>>(mask_pos, ovl, gt_labels,
                                                       gt_bboxes, out_labels,
                                                       out_bboxes, out_fg, out_idx);
  k_rowmax<<<BS * NMAX, 32, 0, stream>>>(align_m, ovl, mask_pos, pos_align, pos_over);
  k_scores<<<(BS * NA + 255) / 256, 256, 0, stream>>>(align_m, mask_pos, pos_align,
                                                      pos_over, out_labels, out_fg,
                                                      out_scores);
}